// GaborRendererCUDA_19456201851646
// MI455X (gfx1250) — compile-verified
//
#include <hip/hip_runtime.h>

// ---------------------------------------------------------------------------
// Gabor atom renderer, gather formulation for MI455X (gfx1250, wave32).
//
//   out[i] = sum_n  amp_n * exp(-0.5*((t_i - tau_n)/sigma_n)^2)
//                  * cos(2*pi*(omega_n*dt + 0.5*gamma_n*dt^2) + phi_n)
//            for |t_i - tau_n| <= 5*sigma_n      (t_i = i / 24000)
//
// Each 256-thread block owns 256 consecutive output samples (1 per thread,
// accumulated in a register -> no atomics, coalesced stores). Atom parameters
// are streamed through LDS in 256-atom chunks using the CDNA5 async
// global->LDS DMA path (global_load_async_to_lds_b32 / s_wait_asynccnt),
// double-buffered so the DMA of chunk c+1 overlaps compute of chunk c.
// Candidate atoms (center within +/-1200 samples of the block) are compacted
// into a 32-byte-per-atom LDS queue (2x ds_load_b128 broadcast per atom on
// the hot path) with a deterministic ballot/prefix-sum so the float
// accumulation order is identical on every replay.
// ---------------------------------------------------------------------------

#define SRATE      24000.0f
#define HALFWIN    1200          // ceil(5 * MAX_SIGMA * SR)
#define BLOCK      256
#define CHUNK      256
#define NARR       6             // amplitude, tau, omega, sigma, phi, gamma
#define NWAVE      (BLOCK / 32)

#if __has_builtin(__builtin_amdgcn_exp2f)
#define FAST_EXP2(x) __builtin_amdgcn_exp2f(x)     // v_exp_f32: 2^x
#else
#define FAST_EXP2(x) exp2f(x)
#endif

#if __has_builtin(__builtin_amdgcn_cosf)
#define COS_REV(x) __builtin_amdgcn_cosf(x)        // v_cos_f32: cos(2*pi*x)
#else
#define COS_REV(x) __cosf((x) * 6.28318530717958647692f)
#endif

#if __has_builtin(__builtin_amdgcn_fractf)
#define FRACT(x) __builtin_amdgcn_fractf(x)        // v_fract_f32: x - floor(x)
#else
#define FRACT(x) ((x) - floorf(x))
#endif

// Generic pointers to LDS carry the LDS byte offset in their low 32 bits
// (flat aperture rule).
__device__ __forceinline__ unsigned lds_addr_of(const void* p) {
  return (unsigned)(unsigned long long)(uintptr_t)p;
}

__global__ __launch_bounds__(BLOCK)
void gabor_gather_kernel(const float* __restrict__ g_amp,
                         const float* __restrict__ g_tau,
                         const float* __restrict__ g_omg,
                         const float* __restrict__ g_sig,
                         const float* __restrict__ g_phi,
                         const float* __restrict__ g_gam,
                         float* __restrict__ out,
                         int N, int num_samples) {
  // Double-buffered async staging area (written only by the async DMA engine).
  __shared__ float stage[2][NARR][CHUNK];
  // Compacted candidate-atom queue: 32B/atom -> two b128 broadcasts per atom.
  //   qd[j][0] = { amp, tau, 5*sigma, 1/sigma }
  //   qd[j][1] = { omega, 0.5*gamma, phi/(2pi), unused }
  __shared__ float4 qd[CHUNK][2];
  __shared__ int    wbase[NWAVE + 1];

  const int   tid = threadIdx.x;
  const int   s   = blockIdx.x * BLOCK + tid;          // my output sample
  const float t   = (float)s * (1.0f / SRATE);
  const float lo  = (float)(blockIdx.x * BLOCK - HALFWIN);
  const float hi  = (float)(blockIdx.x * BLOCK + (BLOCK - 1) + HALFWIN);

  const float* gptr[NARR] = {g_amp, g_tau, g_omg, g_sig, g_phi, g_gam};

  // Issue the 6 async global->LDS loads for one chunk (per-wave DMA,
  // tracked by ASYNCcnt; each lane fetches one element of each array).
  auto issue_prefetch = [&](int buf, int base) {
    int idx = base + tid;
    if (idx >= N) idx = N - 1;                 // clamp: never read OOB
    unsigned voff = (unsigned)idx * 4u;        // GVS mode: saddr + vgpr offset
#pragma unroll
    for (int p = 0; p < NARR; ++p) {
      unsigned laddr = lds_addr_of(&stage[buf][p][tid]);
      asm volatile("global_load_async_to_lds_b32 %0, %1, %2"
                   :: "v"(laddr), "v"(voff), "s"(gptr[p])
                   : "memory");
    }
  };

  float acc = 0.0f;
  int   buf = 0;
  const int nchunks = (N + CHUNK - 1) / CHUNK;

  issue_prefetch(0, 0);

  for (int c = 0; c < nchunks; ++c) {
    // Wait for this chunk's DMA (per-wave; each wave only reads its own lanes
    // of the stage buffer, so no cross-wave barrier is needed for `stage`).
    asm volatile("s_wait_asynccnt 0x0" ::: "memory");

    // Kick off next chunk's DMA into the other buffer -> overlaps compute.
    const int nbase = (c + 1) * CHUNK;
    if (nbase < N) issue_prefetch(buf ^ 1, nbase);

    // ---- scan: which atoms can touch this block's samples? -------------
    const int   i    = c * CHUNK + tid;
    const float taui = stage[buf][1][tid];
    const float ctr  = rintf(taui * SRATE);    // v_rndne (round-half-even)
    const bool  pass = (i < N) && (ctr >= lo) && (ctr <= hi);

    // ---- deterministic compaction (ballot + cross-wave prefix sum) -----
#if __has_builtin(__builtin_amdgcn_ballot_w32)
    const unsigned m = __builtin_amdgcn_ballot_w32(pass);
#else
    const unsigned m = (unsigned)__ballot(pass);
#endif
    const int wv = tid >> 5, ln = tid & 31;
    if (ln == 0) wbase[wv] = __builtin_popcount(m);
    __syncthreads();
    if (tid == 0) {
      int run = 0;
#pragma unroll
      for (int w = 0; w < NWAVE; ++w) { int x = wbase[w]; wbase[w] = run; run += x; }
      wbase[NWAVE] = run;
    }
    __syncthreads();
    if (pass) {
      const int pos = wbase[wv] + __builtin_popcount(m & ((1u << ln) - 1u));
      const float sg = stage[buf][3][tid];
      qd[pos][0] = make_float4(stage[buf][0][tid],            // amp
                               taui,                          // tau
                               5.0f * sg,                     // 5*sigma
                               1.0f / sg);                    // 1/sigma
      qd[pos][1] = make_float4(stage[buf][2][tid],            // omega
                               0.5f * stage[buf][5][tid],     // gamma/2
                               stage[buf][4][tid] * 0.15915493667125702f,
                               0.0f);
    }
    __syncthreads();

    // ---- evaluate candidates: each thread computes its own sample ------
    const int cnt = wbase[NWAVE];
#pragma unroll 2
    for (int j = 0; j < cnt; ++j) {            // uniform addr -> LDS broadcast
      const float4 A = qd[j][0];               // ds_load_b128 (broadcast)
      const float4 B = qd[j][1];               // ds_load_b128 (broadcast)
      const float dt   = t - A.y;
      const float adt  = __builtin_fabsf(dt);
      const float q    = dt * A.w;
      // env = exp(-0.5*q^2) = 2^(q^2 * -0.5*log2(e))  -> one v_exp_f32
      const float env  = FAST_EXP2(q * q * -0.72134751f);
      // phase in revolutions: omega*dt + 0.5*gamma*dt^2 + phi/2pi
      const float ph   = __builtin_fmaf(__builtin_fmaf(B.y, dt, B.x), dt, B.z);
      // one v_fract_f32 keeps the v_cos argument in its accurate range
      const float cz      = COS_REV(FRACT(ph));
      const float contrib = A.x * env * cz;
      acc += (adt <= A.z) ? contrib : 0.0f;
    }
    __syncthreads();                           // queue reused next chunk
    buf ^= 1;
  }

  if (s < num_samples) out[s] = acc;
}

extern "C" void kernel_launch(void* const* d_in, const int* in_sizes, int n_in,
                              void* d_out, int out_size, void* d_ws, size_t ws_size,
                              hipStream_t stream) {
  const float* amp = (const float*)d_in[0];
  const float* tau = (const float*)d_in[1];
  const float* omg = (const float*)d_in[2];
  const float* sig = (const float*)d_in[3];
  const float* phi = (const float*)d_in[4];
  const float* gam = (const float*)d_in[5];
  // d_in[6] is num_samples as a device scalar; out_size equals it.
  const int N  = in_sizes[0];
  const int ns = out_size;
  const int blocks = (ns + BLOCK - 1) / BLOCK;   // 938 blocks for 240000
  gabor_gather_kernel<<<blocks, BLOCK, 0, stream>>>(amp, tau, omg, sig, phi,
                                                    gam, (float*)d_out, N, ns);
}